// GATTreeEncoder_V2_60335700574787
// MI455X (gfx1250) — compile-verified
//
#include <hip/hip_runtime.h>

typedef __attribute__((ext_vector_type(2))) float v2f;
typedef __attribute__((ext_vector_type(8))) float v8f;

// ---- ordered-int mapping so signed atomicMax works for floats of any sign ----
__device__ __forceinline__ int f2oi(float f) {
  int i = __float_as_int(f);
  return (i >= 0) ? i : (i ^ 0x7FFFFFFF);
}
__device__ __forceinline__ float oi2f(int i) {
  return __int_as_float((i >= 0) ? i : (i ^ 0x7FFFFFFF));
}

// =====================================================================
// GEMM: Y[M,Nc] = X[M,K] @ W[K,Nc], fp32, via V_WMMA_F32_16X16X4_F32.
// Each wave32 owns a 16x64 output tile: 4 WMMA accumulators share one
// A fragment per K-step (4x less A traffic, 4 back-to-back WMMAs).
// A frag (16x4): lanes 0-15 hold row M=lane, K={k0,k0+1}; lanes 16-31 same
// rows with K={k0+2,k0+3}.  B frag (4x16) mirrors with N across lanes.
// C/D: VGPR r -> M = r (+8 for lanes>=16), N = lane&15.
// =====================================================================
__global__ void gat_gemm_wmma_f32(const float* __restrict__ X,
                                  const float* __restrict__ W,
                                  float* __restrict__ Y,
                                  int M, int K, int Nc) {
  const int lane = threadIdx.x & 31;
  const int wave = threadIdx.x >> 5;
  const int wpb  = blockDim.x >> 5;
  const int colGroups = Nc >> 6;                     // 64-column groups
  const int gtile = blockIdx.x * wpb + wave;         // wave-uniform
  if (gtile >= (M >> 4) * colGroups) return;         // uniform exit, EXEC stays full
  const int tm  = gtile / colGroups;
  const int tg  = gtile - tm * colGroups;
  const int l15 = lane & 15;
  const int kh  = (lane >> 4) << 1;                  // 0 or 2
  const long long arow = (long long)(tm * 16 + l15) * K;
  const int bcol = tg * 64 + l15;

  v8f acc0 = {0.f, 0.f, 0.f, 0.f, 0.f, 0.f, 0.f, 0.f};
  v8f acc1 = acc0, acc2 = acc0, acc3 = acc0;
  for (int k0 = 0; k0 < K; k0 += 4) {
    // speculative prefetch of A one cache line ahead (global_prefetch_b8)
    __builtin_prefetch(&X[arow + k0 + kh + 32], 0, 1);
    v2f a;
    a.x = X[arow + k0 + kh];
    a.y = X[arow + k0 + kh + 1];
    const float* Wr0 = W + (long long)(k0 + kh) * Nc + bcol;
    const float* Wr1 = Wr0 + Nc;
    v2f b0, b1, b2, b3;
    b0.x = Wr0[0];  b0.y = Wr1[0];
    b1.x = Wr0[16]; b1.y = Wr1[16];
    b2.x = Wr0[32]; b2.y = Wr1[32];
    b3.x = Wr0[48]; b3.y = Wr1[48];
    acc0 = __builtin_amdgcn_wmma_f32_16x16x4_f32(false, a, false, b0, (short)0,
                                                 acc0, false, false);
    acc1 = __builtin_amdgcn_wmma_f32_16x16x4_f32(false, a, false, b1, (short)0,
                                                 acc1, false, false);
    acc2 = __builtin_amdgcn_wmma_f32_16x16x4_f32(false, a, false, b2, (short)0,
                                                 acc2, false, false);
    acc3 = __builtin_amdgcn_wmma_f32_16x16x4_f32(false, a, false, b3, (short)0,
                                                 acc3, false, false);
  }
  const int rbase = tm * 16 + ((lane >> 4) << 3);
#pragma unroll
  for (int r = 0; r < 8; ++r) {
    const long long yr = (long long)(rbase + r) * Nc + bcol;
    Y[yr]      = acc0[r];
    Y[yr + 16] = acc1[r];
    Y[yr + 32] = acc2[r];
    Y[yr + 48] = acc3[r];
  }
}

// ---------------- attention scores: asrc[n,h] = <h[n,h,:], a_src[h,:]> -------
__global__ void gat_att_scores(const float* __restrict__ Hm,
                               const float* __restrict__ aw_s,
                               const float* __restrict__ aw_d,
                               float* __restrict__ asrc, float* __restrict__ adst,
                               int N, int H, int C) {
  int t = blockIdx.x * blockDim.x + threadIdx.x;
  if (t >= N * H) return;
  int h = t % H, n = t / H;
  const float* row = Hm + (long long)n * H * C + (long long)h * C;
  const float* ws  = aw_s + h * C;
  const float* wd  = aw_d + h * C;
  float s = 0.f, d = 0.f;
  for (int c = 0; c < C; ++c) { s += row[c] * ws[c]; d += row[c] * wd[c]; }
  asrc[t] = s;
  adst[t] = d;
}

// ---------------- fills ------------------------------------------------------
__global__ void gat_fill_f32(float* p, float v, long long n) {
  long long t = (long long)blockIdx.x * blockDim.x + threadIdx.x;
  if (t < n) p[t] = v;
}
__global__ void gat_fill_i32(int* p, int v, long long n) {
  long long t = (long long)blockIdx.x * blockDim.x + threadIdx.x;
  if (t < n) p[t] = v;
}

// ---------------- edge pass 1: segment max of leakyrelu(asrc+adst) ----------
__global__ void gat_edge_max(const int* __restrict__ ei, int E, int N, int H,
                             const float* __restrict__ asrc,
                             const float* __restrict__ adst,
                             int* __restrict__ emax) {
  long long t = (long long)blockIdx.x * blockDim.x + threadIdx.x;
  long long total = (long long)(E + N) * H;
  if (t >= total) return;
  int h = (int)(t % H);
  long long e = t / H;
  int s, d;
  if (e < E) { s = ei[e]; d = ei[(long long)E + e]; } else { s = d = (int)(e - E); }
  float v = asrc[(long long)s * H + h] + adst[(long long)d * H + h];
  v = v > 0.f ? v : 0.2f * v;
  atomicMax(&emax[(long long)d * H + h], f2oi(v));
}

// ---------------- edge pass 2: denom[d,h] += exp(e - emax[d,h]) -------------
__global__ void gat_edge_expsum(const int* __restrict__ ei, int E, int N, int H,
                                const float* __restrict__ asrc,
                                const float* __restrict__ adst,
                                const int* __restrict__ emax,
                                float* __restrict__ denom) {
  long long t = (long long)blockIdx.x * blockDim.x + threadIdx.x;
  long long total = (long long)(E + N) * H;
  if (t >= total) return;
  int h = (int)(t % H);
  long long e = t / H;
  int s, d;
  if (e < E) { s = ei[e]; d = ei[(long long)E + e]; } else { s = d = (int)(e - E); }
  long long di = (long long)d * H + h;
  float v = asrc[(long long)s * H + h] + adst[di];
  v = v > 0.f ? v : 0.2f * v;
  atomicAdd(&denom[di], expf(v - oi2f(emax[di])));
}

// ---------------- edge pass 3: out[d,:] += alpha * h[s,:] -------------------
// one block per (edge or self-loop), blockDim = D = H*C
__global__ void gat_edge_scatter(const int* __restrict__ ei, int E, int N,
                                 int H, int C,
                                 const float* __restrict__ asrc,
                                 const float* __restrict__ adst,
                                 const int* __restrict__ emax,
                                 const float* __restrict__ denom,
                                 const float* __restrict__ hsrc,
                                 float* __restrict__ out) {
  __shared__ float salpha[8];                        // H <= 8
  const int e = blockIdx.x;
  const int D = H * C;
  int s, d;
  if (e < E) { s = ei[e]; d = ei[(long long)E + e]; } else { s = d = e - E; }
  if (threadIdx.x < (unsigned)H) {
    int h = threadIdx.x;
    long long di = (long long)d * H + h;
    float v = asrc[(long long)s * H + h] + adst[di];
    v = v > 0.f ? v : 0.2f * v;
    float p = expf(v - oi2f(emax[di]));
    salpha[h] = p / (denom[di] + 1e-16f);
  }
  __syncthreads();
  const int t = threadIdx.x;                         // 0..D-1
  const int h = t / C;
  atomicAdd(&out[(long long)d * D + t], hsrc[(long long)s * D + t] * salpha[h]);
}

// ---------------- fused bias + LayerNorm + residual(+pb) + ELU --------------
// one block per node, blockDim = D (<= 256)
__global__ void gat_ln_res_elu(const float* __restrict__ agg,
                               const float* __restrict__ bias,
                               const float* __restrict__ gw,
                               const float* __restrict__ bw,
                               const float* __restrict__ res,
                               const float* __restrict__ pb,
                               float* __restrict__ out, int D) {
  __shared__ float s1[256];
  __shared__ float s2[256];
  const int n = blockIdx.x, t = threadIdx.x;
  float v = agg[(long long)n * D + t] + bias[t];
  s1[t] = v;
  s2[t] = v * v;
  __syncthreads();
  for (int s = blockDim.x >> 1; s > 0; s >>= 1) {
    if (t < s) { s1[t] += s1[t + s]; s2[t] += s2[t + s]; }
    __syncthreads();
  }
  const float mean = s1[0] / (float)D;
  const float var  = s2[0] / (float)D - mean * mean;
  float y = (v - mean) * rsqrtf(var + 1e-5f) * gw[t] + bw[t]
            + res[(long long)n * D + t] + pb[t];
  out[(long long)n * D + t] = y > 0.f ? y : expm1f(y);
}

// ---------------- global mean pool ------------------------------------------
__global__ void gat_pool_acc(const float* __restrict__ h3,
                             const int* __restrict__ batch,
                             float* __restrict__ out, float* __restrict__ cnt,
                             int N, int D) {
  const int n = blockIdx.x, t = threadIdx.x;
  const int g = batch[n];
  atomicAdd(&out[(long long)g * D + t], h3[(long long)n * D + t]);
  if (t == 0) atomicAdd(&cnt[g], 1.0f);
}
__global__ void gat_pool_div(float* out, const float* cnt, int G, int D) {
  int t = blockIdx.x * blockDim.x + threadIdx.x;
  if (t >= G * D) return;
  out[t] /= fmaxf(cnt[t / D], 1.0f);
}

// =====================================================================
// Host-side per-layer driver (all launches on `stream`, capture-safe)
// =====================================================================
static void gat_run_layer(const float* Xin, int Din, int H, int C,
                          const float* W, const float* aw_s, const float* aw_d,
                          const float* bias, const float* gw, const float* bw,
                          const float* pw, const float* pb,
                          const int* ei, int E, int N,
                          float* bufA, float* bufB, float* bufC,
                          float* asrc, float* adst, int* emax, float* denom,
                          float* Hout, hipStream_t stream) {
  const int D = H * C;
  // dense GEMMs (WMMA): h = Xin@W ; res = Xin@pw
  {
    const int tiles  = (N / 16) * (D / 64);            // 16x64 tile per wave
    const int blocks = (tiles + 7) / 8;                // 8 waves / block
    gat_gemm_wmma_f32<<<blocks, 256, 0, stream>>>(Xin, W,  bufA, N, Din, D);
    gat_gemm_wmma_f32<<<blocks, 256, 0, stream>>>(Xin, pw, bufC, N, Din, D);
  }
  // attention scores
  {
    const int tot = N * H;
    gat_att_scores<<<(tot + 255) / 256, 256, 0, stream>>>(bufA, aw_s, aw_d,
                                                          asrc, adst, N, H, C);
  }
  // init aggregation + softmax state
  {
    const long long nd = (long long)N * D;
    gat_fill_f32<<<(unsigned)((nd + 255) / 256), 256, 0, stream>>>(bufB, 0.f, nd);
    const long long nh = (long long)N * H;
    gat_fill_i32<<<(unsigned)((nh + 255) / 256), 256, 0, stream>>>(emax,
                                                  (int)0x807FFFFF, nh); // f2oi(-inf)
    gat_fill_f32<<<(unsigned)((nh + 255) / 256), 256, 0, stream>>>(denom, 0.f, nh);
  }
  // segment softmax (3 passes over E real edges + N self loops)
  {
    const long long tot = (long long)(E + N) * H;
    const unsigned blks = (unsigned)((tot + 255) / 256);
    gat_edge_max<<<blks, 256, 0, stream>>>(ei, E, N, H, asrc, adst, emax);
    gat_edge_expsum<<<blks, 256, 0, stream>>>(ei, E, N, H, asrc, adst, emax, denom);
    gat_edge_scatter<<<E + N, D, 0, stream>>>(ei, E, N, H, C, asrc, adst,
                                              emax, denom, bufA, bufB);
  }
  // bias + LN + residual + ELU
  gat_ln_res_elu<<<N, D, 0, stream>>>(bufB, bias, gw, bw, bufC, pb, Hout, D);
}

extern "C" void kernel_launch(void* const* d_in, const int* in_sizes, int n_in,
                              void* d_out, int out_size, void* d_ws, size_t ws_size,
                              hipStream_t stream) {
  (void)n_in; (void)ws_size;
  const float* x    = (const float*)d_in[0];
  const int*  ei    = (const int*)d_in[1];
  const int*  batch = (const int*)d_in[2];
  const float* W1 = (const float*)d_in[3],  *as1 = (const float*)d_in[4],
             *ad1 = (const float*)d_in[5],  *b1  = (const float*)d_in[6],
             *g1  = (const float*)d_in[7],  *be1 = (const float*)d_in[8],
             *pw1 = (const float*)d_in[9],  *pb1 = (const float*)d_in[10];
  const float* W2 = (const float*)d_in[11], *as2 = (const float*)d_in[12],
             *ad2 = (const float*)d_in[13], *b2  = (const float*)d_in[14],
             *g2  = (const float*)d_in[15], *be2 = (const float*)d_in[16],
             *pw2 = (const float*)d_in[17], *pb2 = (const float*)d_in[18];
  const float* W3 = (const float*)d_in[19], *as3 = (const float*)d_in[20],
             *ad3 = (const float*)d_in[21], *b3  = (const float*)d_in[22],
             *g3  = (const float*)d_in[23], *be3 = (const float*)d_in[24],
             *pw3 = (const float*)d_in[25], *pb3 = (const float*)d_in[26];

  const int N = in_sizes[0] / 128;       // 50000
  const int E = in_sizes[1] / 2;         // 800000
  const int G = out_size / 128;          // 512

  // workspace layout (floats)
  float* ws = (float*)d_ws;
  size_t o = 0;
  float* bufA  = ws + o; o += (size_t)N * 256;   // pre-aggregation features
  float* bufB  = ws + o; o += (size_t)N * 256;   // aggregated output
  float* bufC  = ws + o; o += (size_t)N * 256;   // residual GEMM
  float* Hp0   = ws + o; o += (size_t)N * 256;   // layer activations (ping)
  float* Hp1   = ws + o; o += (size_t)N * 256;   // layer activations (pong)
  float* asrc  = ws + o; o += (size_t)N * 8;
  float* adst  = ws + o; o += (size_t)N * 8;
  int*   emax  = (int*)(ws + o); o += (size_t)N * 8;
  float* denom = ws + o; o += (size_t)N * 8;
  float* cnt   = ws + o; o += (size_t)G;

  // Layer 1: 128 -> 8 heads x 32 (concat => 256)
  gat_run_layer(x,   128, 8, 32,  W1, as1, ad1, b1, g1, be1, pw1, pb1,
                ei, E, N, bufA, bufB, bufC, asrc, adst, emax, denom, Hp0, stream);
  // Layer 2: 256 -> 4 heads x 32 (concat => 128)
  gat_run_layer(Hp0, 256, 4, 32,  W2, as2, ad2, b2, g2, be2, pw2, pb2,
                ei, E, N, bufA, bufB, bufC, asrc, adst, emax, denom, Hp1, stream);
  // Layer 3: 128 -> 1 head x 128 (mean over 1 head == identity)
  gat_run_layer(Hp1, 128, 1, 128, W3, as3, ad3, b3, g3, be3, pw3, pb3,
                ei, E, N, bufA, bufB, bufC, asrc, adst, emax, denom, Hp0, stream);

  // global mean pool per graph
  float* out = (float*)d_out;
  {
    const long long gd = (long long)G * 128;
    gat_fill_f32<<<(unsigned)((gd + 255) / 256), 256, 0, stream>>>(out, 0.f, gd);
    gat_fill_f32<<<(unsigned)((G + 255) / 256), 256, 0, stream>>>(cnt, 0.f, G);
    gat_pool_acc<<<N, 128, 0, stream>>>(Hp0, batch, out, cnt, N, 128);
    gat_pool_div<<<(G * 128 + 255) / 256, 256, 0, stream>>>(out, cnt, G, 128);
  }
}